// GraphConv_76398878261701
// MI455X (gfx1250) — compile-verified
//
#include <hip/hip_runtime.h>

typedef __attribute__((ext_vector_type(2))) float v2f;
typedef __attribute__((ext_vector_type(8))) float v8f;

#define B_ 4
#define C_ 64
#define N_ 16384
#define K_ 32
#define LDSS 68   // LDS row stride (floats): 68 % 64 == 4 -> conflict-free wmma operand fetch

// ---------------------------------------------------------------------------
// Kernel 0: transpose x [B,C,N] -> xT [B,N,C] so each neighbor gather is one
// contiguous 256B line (float2 per lane, wave32-coalesced).
// ---------------------------------------------------------------------------
__global__ __launch_bounds__(256) void k_transpose(const float* __restrict__ x,
                                                   float* __restrict__ xT) {
  __shared__ float tile[32][33];
  const int b  = blockIdx.z;
  const int c0 = blockIdx.y * 32;
  const int n0 = blockIdx.x * 32;
  const float* xb = x + (size_t)b * C_ * N_;
  float* xTb = xT + (size_t)b * N_ * C_;
  for (int i = threadIdx.y; i < 32; i += 8)
    tile[i][threadIdx.x] = xb[(size_t)(c0 + i) * N_ + n0 + threadIdx.x];
  __syncthreads();
  for (int i = threadIdx.y; i < 32; i += 8)
    xTb[(size_t)(n0 + i) * C_ + c0 + threadIdx.x] = tile[threadIdx.x][i];
}

// ---------------------------------------------------------------------------
// Kernel 1: fused gather-mean + 1x1 conv via V_WMMA_F32_16X16X4_F32.
// One wave32 per 16-point tile; 8 waves (256 thr) per block.
//   agg tile  : 16 points x 64 channels (LDS, stride 68)
//   W         : 64x64 preloaded to LDS (stride 68)
//   D = W(16x4-tiles) x agg^T(4x16) accumulated over 16 K-steps, 4 o-tiles
// ---------------------------------------------------------------------------
__global__ __launch_bounds__(256) void k_gather_gemm(
    const float* __restrict__ xT, const long long* __restrict__ edges,
    const float* __restrict__ W, float* __restrict__ h) {
  __shared__ float Wl[C_ * LDSS];          // 17.4 KB
  __shared__ float agg[8][16 * LDSS];      // 34.8 KB

  const int tid = threadIdx.x;
  for (int i = tid; i < C_ * C_; i += 256) {
    const int r = i >> 6, c = i & 63;
    Wl[r * LDSS + c] = W[i];
  }
  __syncthreads();

  const int wave = tid >> 5;
  const int lane = tid & 31;
  const int tileId = blockIdx.x * 8 + wave;
  const int b  = tileId / (N_ / 16);
  const int n0 = (tileId % (N_ / 16)) * 16;

  const float* xb = xT + (size_t)b * N_ * C_;
  float* aggw = agg[wave];

  // ---- gather + neighbor mean: lane owns channels {2*lane, 2*lane+1} ----
  for (int p = 0; p < 16; ++p) {
    const long long e64 = edges[((size_t)b * N_ + (n0 + p)) * K_ + lane];
    const int ei = (int)e64;
    float ax = 0.f, ay = 0.f;
#pragma unroll
    for (int k = 0; k < 32; ++k) {
      const int s = __builtin_amdgcn_readlane(ei, k);   // uniform SGPR index
      const float* src = xb + (size_t)s * C_ + 2 * lane;
      ax += src[0];
      ay += src[1];
    }
    aggw[p * LDSS + 2 * lane]     = ax * (1.f / 32.f);
    aggw[p * LDSS + 2 * lane + 1] = ay * (1.f / 32.f);
  }
  __syncthreads();

  // ---- WMMA: D[o-tile 16 x n 16] += A(W 16x4) * B(agg^T 4x16) ----
  // ISA layout: lanes 0-15 hold K=0,1 (vgpr0,1); lanes 16-31 hold K=2,3.
  const int half = lane >> 4;
  const int lrow = lane & 15;
  v8f acc0 = {}, acc1 = {}, acc2 = {}, acc3 = {};
#pragma unroll
  for (int c0 = 0; c0 < C_; c0 += 4) {
    const int col = c0 + 2 * half;
    const float* ar = aggw + lrow * LDSS + col;
    v2f bv = {ar[0], ar[1]};                          // B[k][n'] = agg[n'][c0+k]
    const float* w0 = Wl + (0 * 16 + lrow) * LDSS + col;
    const float* w1 = Wl + (1 * 16 + lrow) * LDSS + col;
    const float* w2 = Wl + (2 * 16 + lrow) * LDSS + col;
    const float* w3 = Wl + (3 * 16 + lrow) * LDSS + col;
    v2f a0 = {w0[0], w0[1]};
    v2f a1 = {w1[0], w1[1]};
    v2f a2 = {w2[0], w2[1]};
    v2f a3 = {w3[0], w3[1]};
    acc0 = __builtin_amdgcn_wmma_f32_16x16x4_f32(false, a0, false, bv, (short)0, acc0, false, false);
    acc1 = __builtin_amdgcn_wmma_f32_16x16x4_f32(false, a1, false, bv, (short)0, acc1, false, false);
    acc2 = __builtin_amdgcn_wmma_f32_16x16x4_f32(false, a2, false, bv, (short)0, acc2, false, false);
    acc3 = __builtin_amdgcn_wmma_f32_16x16x4_f32(false, a3, false, bv, (short)0, acc3, false, false);
  }

  // ---- store D: vgpr r of lane l -> (o = otile*16 + r + 8*half, n = n0+lrow)
  float* hb = h + (size_t)b * C_ * N_ + n0 + lrow;
#pragma unroll
  for (int r = 0; r < 8; ++r) {
    const int ro = r + 8 * half;
    hb[(size_t)(0 * 16 + ro) * N_] = acc0[r];
    hb[(size_t)(1 * 16 + ro) * N_] = acc1[r];
    hb[(size_t)(2 * 16 + ro) * N_] = acc2[r];
    hb[(size_t)(3 * 16 + ro) * N_] = acc3[r];
  }
}

// ---------------------------------------------------------------------------
// Kernel 2: per-channel batch stats over (B, N); emit fused scale/shift.
// ---------------------------------------------------------------------------
__global__ __launch_bounds__(256) void k_bnstats(const float* __restrict__ h,
                                                 const float* __restrict__ gamma,
                                                 const float* __restrict__ beta,
                                                 float* __restrict__ sc,
                                                 float* __restrict__ sh) {
  __shared__ float ssum[256], ssq[256];
  const int o = blockIdx.x;
  float s = 0.f, q = 0.f;
  for (int b = 0; b < B_; ++b) {
    const float* row = h + ((size_t)b * C_ + o) * N_;
    for (int n = threadIdx.x; n < N_; n += 256) {
      const float v = row[n];
      s += v;
      q += v * v;
    }
  }
  ssum[threadIdx.x] = s;
  ssq[threadIdx.x]  = q;
  __syncthreads();
  for (int st = 128; st > 0; st >>= 1) {
    if ((int)threadIdx.x < st) {
      ssum[threadIdx.x] += ssum[threadIdx.x + st];
      ssq[threadIdx.x]  += ssq[threadIdx.x + st];
    }
    __syncthreads();
  }
  if (threadIdx.x == 0) {
    const float inv = 1.f / (float)(B_ * N_);
    const float mu  = ssum[0] * inv;
    const float var = ssq[0] * inv - mu * mu;      // biased variance
    const float rs  = rsqrtf(var + 1e-5f);
    const float scale = gamma[o] * rs;
    sc[o] = scale;
    sh[o] = beta[o] - mu * scale;
  }
}

// ---------------------------------------------------------------------------
// Kernel 3: y = leaky( scale[o]*h + shift[o] ), float4-vectorized.
// ---------------------------------------------------------------------------
__global__ __launch_bounds__(256) void k_bnact(const float* __restrict__ h,
                                               const float* __restrict__ sc,
                                               const float* __restrict__ sh,
                                               float* __restrict__ out) {
  const size_t i = ((size_t)blockIdx.x * 256 + threadIdx.x) * 4;
  const int o = (int)((i / N_) & (C_ - 1));
  const float4 v = *(const float4*)(h + i);
  const float scale = sc[o], shift = sh[o];
  float4 y;
  y.x = fmaf(v.x, scale, shift);
  y.y = fmaf(v.y, scale, shift);
  y.z = fmaf(v.z, scale, shift);
  y.w = fmaf(v.w, scale, shift);
  y.x = y.x > 0.f ? y.x : 0.2f * y.x;
  y.y = y.y > 0.f ? y.y : 0.2f * y.y;
  y.z = y.z > 0.f ? y.z : 0.2f * y.z;
  y.w = y.w > 0.f ? y.w : 0.2f * y.w;
  *(float4*)(out + i) = y;
}

// ---------------------------------------------------------------------------
extern "C" void kernel_launch(void* const* d_in, const int* in_sizes, int n_in,
                              void* d_out, int out_size, void* d_ws, size_t ws_size,
                              hipStream_t stream) {
  const float*     x     = (const float*)d_in[0];
  const long long* edges = (const long long*)d_in[1];
  const float*     W     = (const float*)d_in[2];
  const float*     gamma = (const float*)d_in[3];
  const float*     beta  = (const float*)d_in[4];
  float* out = (float*)d_out;

  char* ws = (char*)d_ws;
  const size_t elems = (size_t)B_ * C_ * N_;        // 4,194,304
  float* xT = (float*)ws;                           // 16 MB
  float* h  = (float*)(ws + elems * sizeof(float)); // 16 MB
  float* sc = (float*)(ws + 2 * elems * sizeof(float));
  float* sh = sc + C_;

  dim3 tgrid(N_ / 32, C_ / 32, B_);
  k_transpose<<<tgrid, dim3(32, 8), 0, stream>>>(x, xT);

  const int tiles = B_ * (N_ / 16);                 // 4096 wave-tiles
  k_gather_gemm<<<tiles / 8, 256, 0, stream>>>(xT, edges, W, h);

  k_bnstats<<<C_, 256, 0, stream>>>(h, gamma, beta, sc, sh);

  k_bnact<<<(int)(elems / 4 / 256), 256, 0, stream>>>(h, sc, sh, out);
}